// DeepseekV2MoE_17583596109835
// MI455X (gfx1250) — compile-verified
//
#include <hip/hip_runtime.h>
#include <hip/hip_bf16.h>

// ---------------- problem constants ----------------
#define T_TOK   1024
#define HID     2048
#define NEXP    16
#define TOPK    4
#define NGROUP  4
#define TOPKG   2
#define IMOE    1408
#define SHI     2816        // SI = I * n_shared
#define ROUTED_SCALE 1.0f

typedef __attribute__((ext_vector_type(2))) float v2f;
typedef __attribute__((ext_vector_type(8))) float v8f;

__device__ __forceinline__ v8f wmma4(v2f a, v2f b, v8f c) {
  // D = A(16x4 f32) * B(4x16 f32) + C(16x16 f32)
  return __builtin_amdgcn_wmma_f32_16x16x4_f32(
      false, a, false, b, (short)0, c, false, false);
}

__device__ __forceinline__ v8f vzero8() {
  v8f z;
  #pragma unroll
  for (int i = 0; i < 8; i++) z[i] = 0.0f;
  return z;
}

__device__ __forceinline__ float silu(float g) { return g / (1.0f + __expf(-g)); }

// --- CDNA5 async global->LDS copy (ASYNCcnt-tracked, bypasses VGPRs) -------
__device__ __forceinline__ unsigned lds_off(const void* p) {
  // low 32 bits of the flat address of an LDS object == wave-relative offset
  return (unsigned)(size_t)p;
}

__device__ __forceinline__ void async_copy16(unsigned lds_byte, unsigned gbyte,
                                             const float* sbase) {
  // GVS mode: saddr = uniform 64-bit base, vaddr = per-lane 32-bit byte offset
  asm volatile("global_load_async_to_lds_b128 %0, %1, %2"
               :: "v"(lds_byte), "v"(gbyte), "s"(sbase) : "memory");
}

__device__ __forceinline__ void wait_async0() {
#if __has_builtin(__builtin_amdgcn_s_wait_asynccnt)
  __builtin_amdgcn_s_wait_asynccnt(0);
#else
  asm volatile("s_wait_asynccnt 0x0" ::: "memory");
#endif
}

// ---------------------------------------------------------------------------
// Kernel 0: router — logits, softmax, grouped top-k, dense combine weights
// ---------------------------------------------------------------------------
__global__ __launch_bounds__(128)
void router_kernel(const float* __restrict__ x,
                   const float* __restrict__ gw,   // [E, H]
                   float* __restrict__ cw)         // [T, E]
{
  const int lane = threadIdx.x & 31;
  const int t = blockIdx.x * 4 + (threadIdx.x >> 5);

  float acc[NEXP];
  #pragma unroll
  for (int e = 0; e < NEXP; e++) acc[e] = 0.0f;

  for (int h = lane; h < HID; h += 32) {
    const float xv = x[t * HID + h];
    #pragma unroll
    for (int e = 0; e < NEXP; e++) acc[e] += xv * gw[e * HID + h];
  }
  #pragma unroll
  for (int e = 0; e < NEXP; e++) {
    float v = acc[e];
    #pragma unroll
    for (int off = 16; off > 0; off >>= 1) v += __shfl_xor(v, off, 32);
    acc[e] = v;
  }

  if (lane == 0) {
    float mx = acc[0];
    #pragma unroll
    for (int e = 1; e < NEXP; e++) mx = fmaxf(mx, acc[e]);
    float sc[NEXP]; float s = 0.0f;
    #pragma unroll
    for (int e = 0; e < NEXP; e++) { sc[e] = __expf(acc[e] - mx); s += sc[e]; }
    #pragma unroll
    for (int e = 0; e < NEXP; e++) sc[e] /= s;

    float gs[NGROUP];
    #pragma unroll
    for (int g = 0; g < NGROUP; g++) {
      float m = sc[g * 4];
      #pragma unroll
      for (int j = 1; j < 4; j++) m = fmaxf(m, sc[g * 4 + j]);
      gs[g] = m;
    }
    bool gmask[NGROUP];
    #pragma unroll
    for (int g = 0; g < NGROUP; g++) gmask[g] = false;
    for (int it = 0; it < TOPKG; it++) {
      int best = -1; float bv = -1e30f;
      #pragma unroll
      for (int g = 0; g < NGROUP; g++)
        if (!gmask[g] && gs[g] > bv) { bv = gs[g]; best = g; }
      gmask[best] = true;
    }
    float tmp[NEXP];
    #pragma unroll
    for (int e = 0; e < NEXP; e++) tmp[e] = gmask[e >> 2] ? sc[e] : 0.0f;
    int id[TOPK]; float wv[TOPK]; float wsum = 0.0f;
    for (int k = 0; k < TOPK; k++) {
      int best = 0; float bv = -1e30f;
      #pragma unroll
      for (int e = 0; e < NEXP; e++)
        if (tmp[e] > bv) { bv = tmp[e]; best = e; }
      id[k] = best; wv[k] = bv; wsum += bv; tmp[best] = -1.0f;
    }
    float row[NEXP];
    #pragma unroll
    for (int e = 0; e < NEXP; e++) row[e] = 0.0f;
    #pragma unroll
    for (int k = 0; k < TOPK; k++) row[id[k]] = wv[k] / wsum;
    #pragma unroll
    for (int e = 0; e < NEXP; e++) cw[t * NEXP + e] = row[e];
  }
}

// ---------------------------------------------------------------------------
// Kernel 1: deterministic per-expert token compaction (one wave per expert)
// ---------------------------------------------------------------------------
__global__ __launch_bounds__(32)
void compact_kernel(const float* __restrict__ cw,
                    int* __restrict__ list, float* __restrict__ wts,
                    int* __restrict__ cnt)
{
  const int e = blockIdx.x;
  const int lane = threadIdx.x;
  int base = 0;
  for (int c = 0; c < T_TOK / 32; c++) {
    const int t = c * 32 + lane;
    const float w = cw[t * NEXP + e];
    const bool act = (w > 0.0f);
    const unsigned m32 = (unsigned)__ballot(act);
    const int prefix = __popc(m32 & ((1u << lane) - 1u));
    if (act) {
      list[e * T_TOK + base + prefix] = t;
      wts [e * T_TOK + base + prefix] = w;
    }
    base += __popc(m32);
  }
  if (lane == 0) cnt[e] = base;
}

// ---------------------------------------------------------------------------
// GEMM tile geometry
//   block = 256 threads (8 waves); block tile M=128, N=64; K-tile = 32
//   A staged [m][k] (stride LDA), B staged UN-transposed [k][n] (stride LDBN)
//   staging via global_load_async_to_lds_b128, double-buffered
// ---------------------------------------------------------------------------
#define BM   128
#define BN   64
#define KT   32
#define LDA  36            // A row stride (f32): 144 B rows, 16B aligned
#define LDBN 80            // B row stride (f32): 320 B rows, 16B aligned
#define A_SZ (BM * LDA)    // 4608 f32
#define B_SZ (KT * LDBN)   // 2560 f32

// ---------------------------------------------------------------------------
// Kernel 2: shared experts gate_up GEMM + SiluAndMul epilogue
// layout: A0 | A1 | Bg0 | Bg1 | Bu0 | Bu1
// ---------------------------------------------------------------------------
__global__ __launch_bounds__(256)
void shared_gateup_kernel(const float* __restrict__ x,
                          const float* __restrict__ w,    // [H, 2*SHI]
                          float* __restrict__ act)        // [T, SHI]
{
  __shared__ float smem[2 * A_SZ + 4 * B_SZ];             // 19456 f32

  const int m0 = blockIdx.x * BM;
  const int n0 = blockIdx.y * BN;
  const int tid = threadIdx.x;
  const int lane = tid & 31;
  const int wv = tid >> 5;
  const int WROW = 2 * SHI;

  const int ar = tid >> 3, ac = (tid & 7) * 4;    // A staging coords
  const int bk = tid >> 4, bc = (tid & 15) * 4;   // B staging coords
  const unsigned sbase = lds_off(smem);

  auto stage = [&](int buf, int k0) {
    const unsigned aoff = sbase + (unsigned)(buf * A_SZ * 4);
    const unsigned goff = sbase + (unsigned)((2 * A_SZ + buf * B_SZ) * 4);
    const unsigned uoff = sbase + (unsigned)((2 * A_SZ + (2 + buf) * B_SZ) * 4);
    #pragma unroll
    for (int i = 0; i < 4; i++) {
      const int row = ar + 32 * i;
      async_copy16(aoff + (unsigned)((row * LDA + ac) * 4),
                   (unsigned)(((m0 + row) * HID + k0 + ac) * 4), x);
    }
    #pragma unroll
    for (int i = 0; i < 2; i++) {
      const int kk = bk + 16 * i;
      const unsigned l = (unsigned)((kk * LDBN + bc) * 4);
      async_copy16(goff + l, (unsigned)(((k0 + kk) * WROW + n0 + bc) * 4), w);
      async_copy16(uoff + l, (unsigned)(((k0 + kk) * WROW + SHI + n0 + bc) * 4), w);
    }
  };

  v8f accg[4], accu[4];
  #pragma unroll
  for (int i = 0; i < 4; i++) { accg[i] = vzero8(); accu[i] = vzero8(); }

  stage(0, 0);
  wait_async0();
  __syncthreads();

  const int mrow  = (wv * 16 + (lane & 15)) * LDA;
  const int khalf = (lane >> 4) * 2;
  const int nbl   = lane & 15;

  for (int k0 = 0; k0 < HID; k0 += KT) {
    const int buf = (k0 / KT) & 1;
    if (k0 + KT < HID) stage(buf ^ 1, k0 + KT);
    const float* A  = smem + buf * A_SZ;
    const float* Bg = smem + 2 * A_SZ + buf * B_SZ;
    const float* Bu = smem + 2 * A_SZ + (2 + buf) * B_SZ;
    #pragma unroll
    for (int ks = 0; ks < KT; ks += 4) {
      const int ka = ks + khalf;
      v2f a; a.x = A[mrow + ka]; a.y = A[mrow + ka + 1];
      #pragma unroll
      for (int nt = 0; nt < 4; nt++) {
        const int nb = ka * LDBN + nt * 16 + nbl;
        v2f bg; bg.x = Bg[nb]; bg.y = Bg[nb + LDBN];
        v2f bu; bu.x = Bu[nb]; bu.y = Bu[nb + LDBN];
        accg[nt] = wmma4(a, bg, accg[nt]);
        accu[nt] = wmma4(a, bu, accu[nt]);
      }
    }
    wait_async0();
    __syncthreads();
  }

  const int rbase = wv * 16 + (lane >> 4) * 8;
  const int col = lane & 15;
  #pragma unroll
  for (int nt = 0; nt < 4; nt++) {
    const int n = n0 + nt * 16 + col;
    #pragma unroll
    for (int r = 0; r < 8; r++) {
      const float g = accg[nt][r], u = accu[nt][r];
      act[(size_t)(m0 + rbase + r) * SHI + n] = silu(g) * u;
    }
  }
}

// ---------------------------------------------------------------------------
// Kernel 3: shared experts down-proj, writes (initializes) out[T, H]
// layout: A0 | A1 | B0 | B1
// ---------------------------------------------------------------------------
__global__ __launch_bounds__(256)
void shared_down_kernel(const float* __restrict__ act,  // [T, SHI]
                        const float* __restrict__ w,    // [SHI, H]
                        float* __restrict__ out)        // [T, H]
{
  __shared__ float smem[2 * A_SZ + 2 * B_SZ];           // 14336 f32

  const int m0 = blockIdx.x * BM;
  const int n0 = blockIdx.y * BN;
  const int tid = threadIdx.x;
  const int lane = tid & 31;
  const int wv = tid >> 5;

  const int ar = tid >> 3, ac = (tid & 7) * 4;
  const int bk = tid >> 4, bc = (tid & 15) * 4;
  const unsigned sbase = lds_off(smem);

  auto stage = [&](int buf, int k0) {
    const unsigned aoff = sbase + (unsigned)(buf * A_SZ * 4);
    const unsigned boff = sbase + (unsigned)((2 * A_SZ + buf * B_SZ) * 4);
    #pragma unroll
    for (int i = 0; i < 4; i++) {
      const int row = ar + 32 * i;
      async_copy16(aoff + (unsigned)((row * LDA + ac) * 4),
                   (unsigned)(((m0 + row) * SHI + k0 + ac) * 4), act);
    }
    #pragma unroll
    for (int i = 0; i < 2; i++) {
      const int kk = bk + 16 * i;
      async_copy16(boff + (unsigned)((kk * LDBN + bc) * 4),
                   (unsigned)(((k0 + kk) * HID + n0 + bc) * 4), w);
    }
  };

  v8f acc[4];
  #pragma unroll
  for (int i = 0; i < 4; i++) acc[i] = vzero8();

  stage(0, 0);
  wait_async0();
  __syncthreads();

  const int mrow  = (wv * 16 + (lane & 15)) * LDA;
  const int khalf = (lane >> 4) * 2;
  const int nbl   = lane & 15;

  for (int k0 = 0; k0 < SHI; k0 += KT) {
    const int buf = (k0 / KT) & 1;
    if (k0 + KT < SHI) stage(buf ^ 1, k0 + KT);
    const float* A = smem + buf * A_SZ;
    const float* B = smem + 2 * A_SZ + buf * B_SZ;
    #pragma unroll
    for (int ks = 0; ks < KT; ks += 4) {
      const int ka = ks + khalf;
      v2f a; a.x = A[mrow + ka]; a.y = A[mrow + ka + 1];
      #pragma unroll
      for (int nt = 0; nt < 4; nt++) {
        const int nb = ka * LDBN + nt * 16 + nbl;
        v2f b; b.x = B[nb]; b.y = B[nb + LDBN];
        acc[nt] = wmma4(a, b, acc[nt]);
      }
    }
    wait_async0();
    __syncthreads();
  }

  const int rbase = wv * 16 + (lane >> 4) * 8;
  const int col = lane & 15;
  #pragma unroll
  for (int nt = 0; nt < 4; nt++) {
    const int n = n0 + nt * 16 + col;
    #pragma unroll
    for (int r = 0; r < 8; r++)
      out[(size_t)(m0 + rbase + r) * HID + n] = acc[nt][r];
  }
}

// ---------------------------------------------------------------------------
// Kernel 4: routed experts, fused per (m-tile, I-slice, expert)
//   phase1: gathered gate_up GEMM (K=H) + SiluAndMul -> LDS act [128 x 64]
//   phase2: act @ w_down[e][i-slice, :] * combine weight, atomic accumulate
// phase1 layout: A0 | A1 | Bg0 | Bg1 | Bu0 | Bu1      (19456 f32)
// phase2 layout: sAct (128*LDACT=8704) | Bd0 | Bd1    (18944 f32)
// ---------------------------------------------------------------------------
#define ISL    64
#define LDACT  68
#define ACT_SZ (BM * LDACT)   // 8704 f32
#define BD_SZ  (ISL * LDBN)   // 5120 f32

__global__ __launch_bounds__(256)
void routed_fused_kernel(const float* __restrict__ x,         // [T, H]
                         const float* __restrict__ wgu,       // [E, H, 2*I]
                         const float* __restrict__ wd,        // [E, I, H]
                         const int*   __restrict__ list,      // [E, T]
                         const float* __restrict__ wts,       // [E, T]
                         const int*   __restrict__ cnt,       // [E]
                         float* __restrict__ out)             // [T, H]
{
  __shared__ float smem[2 * A_SZ + 4 * B_SZ];   // 19456 f32 (covers both phases)
  __shared__ int   s_tok[BM];
  __shared__ float s_wt [BM];

  const int e  = blockIdx.z;
  const int m0 = blockIdx.x * BM;
  const int i0 = blockIdx.y * ISL;
  const int cntE = cnt[e];
  if (m0 >= cntE) return;

  const int tid = threadIdx.x;
  const int lane = tid & 31;
  const int wv = tid >> 5;

  if (tid < BM) {
    const int idx = m0 + tid;
    const bool ok = (idx < cntE);
    s_tok[tid] = ok ? list[e * T_TOK + idx] : -1;
    s_wt [tid] = ok ? wts [e * T_TOK + idx] : 0.0f;
  }
  __syncthreads();

  const int ar = tid >> 3, ac = (tid & 7) * 4;
  const int bk = tid >> 4, bc = (tid & 15) * 4;
  const unsigned sbase = lds_off(smem);

  int trow[4];
  #pragma unroll
  for (int i = 0; i < 4; i++) {
    const int tk = s_tok[ar + 32 * i];
    trow[i] = (tk < 0) ? 0 : tk;
  }

  const float* wgub = wgu + (size_t)e * HID * (2 * IMOE);
  const float* wdb  = wd  + (size_t)e * IMOE * HID;
  const int WROW = 2 * IMOE;

  auto stage1 = [&](int buf, int k0) {
    const unsigned aoff = sbase + (unsigned)(buf * A_SZ * 4);
    const unsigned goff = sbase + (unsigned)((2 * A_SZ + buf * B_SZ) * 4);
    const unsigned uoff = sbase + (unsigned)((2 * A_SZ + (2 + buf) * B_SZ) * 4);
    #pragma unroll
    for (int i = 0; i < 4; i++) {
      const int row = ar + 32 * i;
      async_copy16(aoff + (unsigned)((row * LDA + ac) * 4),
                   (unsigned)((trow[i] * HID + k0 + ac) * 4), x);
    }
    #pragma unroll
    for (int i = 0; i < 2; i++) {
      const int kk = bk + 16 * i;
      const unsigned l = (unsigned)((kk * LDBN + bc) * 4);
      async_copy16(goff + l, (unsigned)(((k0 + kk) * WROW + i0 + bc) * 4), wgub);
      async_copy16(uoff + l, (unsigned)(((k0 + kk) * WROW + IMOE + i0 + bc) * 4), wgub);
    }
  };

  // ---- phase 1 ----
  v8f accg[4], accu[4];
  #pragma unroll
  for (int i = 0; i < 4; i++) { accg[i] = vzero8(); accu[i] = vzero8(); }

  stage1(0, 0);
  wait_async0();
  __syncthreads();

  const int mrow  = (wv * 16 + (lane & 15)) * LDA;
  const int khalf = (lane >> 4) * 2;
  const int nbl   = lane & 15;

  for (int k0 = 0; k0 < HID; k0 += KT) {
    const int buf = (k0 / KT) & 1;
    if (k0 + KT < HID) stage1(buf ^ 1, k0 + KT);
    const float* A  = smem + buf * A_SZ;
    const float* Bg = smem + 2 * A_SZ + buf * B_SZ;
    const float* Bu = smem + 2 * A_SZ + (2 + buf) * B_SZ;
    #pragma unroll
    for (int ks = 0; ks < KT; ks += 4) {
      const int ka = ks + khalf;
      v2f a; a.x = A[mrow + ka]; a.y = A[mrow + ka + 1];
      #pragma unroll
      for (int nt = 0; nt < 4; nt++) {
        const int nb = ka * LDBN + nt * 16 + nbl;
        v2f bg; bg.x = Bg[nb]; bg.y = Bg[nb + LDBN];
        v2f bu; bu.x = Bu[nb]; bu.y = Bu[nb + LDBN];
        accg[nt] = wmma4(a, bg, accg[nt]);
        accu[nt] = wmma4(a, bu, accu[nt]);
      }
    }
    wait_async0();
    __syncthreads();
  }

  // SiluAndMul -> LDS act tile (overwrites phase-1 buffers; reads complete)
  {
    float* sAct = smem;
    const int rbase = wv * 16 + (lane >> 4) * 8;
    const int col = lane & 15;
    #pragma unroll
    for (int nt = 0; nt < 4; nt++) {
      #pragma unroll
      for (int r = 0; r < 8; r++) {
        const float g = accg[nt][r], u = accu[nt][r];
        sAct[(rbase + r) * LDACT + nt * 16 + col] = silu(g) * u;
      }
    }
  }
  __syncthreads();

  // ---- phase 2 ----
  auto stage2 = [&](int buf, int n0h) {
    const unsigned doff = sbase + (unsigned)((ACT_SZ + buf * BD_SZ) * 4);
    #pragma unroll
    for (int i = 0; i < 4; i++) {
      const int kk = bk + 16 * i;   // 0..63 over the I-slice
      async_copy16(doff + (unsigned)((kk * LDBN + bc) * 4),
                   (unsigned)(((i0 + kk) * HID + n0h + bc) * 4), wdb);
    }
  };

  stage2(0, 0);
  wait_async0();
  __syncthreads();

  const float* sAct = smem;
  const int mrow2 = (wv * 16 + (lane & 15)) * LDACT;
  const int rbase = wv * 16 + (lane >> 4) * 8;
  const int col = lane & 15;

  for (int n0h = 0; n0h < HID; n0h += ISL) {
    const int buf = (n0h / ISL) & 1;
    if (n0h + ISL < HID) stage2(buf ^ 1, n0h + ISL);
    const float* Bd = smem + ACT_SZ + buf * BD_SZ;

    v8f acc2[4];
    #pragma unroll
    for (int i = 0; i < 4; i++) acc2[i] = vzero8();

    #pragma unroll
    for (int ks = 0; ks < ISL; ks += 4) {
      const int ka = ks + khalf;
      v2f a; a.x = sAct[mrow2 + ka]; a.y = sAct[mrow2 + ka + 1];
      #pragma unroll
      for (int nt = 0; nt < 4; nt++) {
        const int nb = ka * LDBN + nt * 16 + nbl;
        v2f b; b.x = Bd[nb]; b.y = Bd[nb + LDBN];
        acc2[nt] = wmma4(a, b, acc2[nt]);
      }
    }

    #pragma unroll
    for (int nt = 0; nt < 4; nt++) {
      const int n = n0h + nt * 16 + col;
      #pragma unroll
      for (int r = 0; r < 8; r++) {
        const int rr = rbase + r;
        const int tk = s_tok[rr];
        if (tk >= 0) {
          const float v = acc2[nt][r] * s_wt[rr] * ROUTED_SCALE;
          atomicAdd(&out[(size_t)tk * HID + n], v);
        }
      }
    }
    wait_async0();
    __syncthreads();
  }
}

// ---------------------------------------------------------------------------
// launcher
// ---------------------------------------------------------------------------
extern "C" void kernel_launch(void* const* d_in, const int* in_sizes, int n_in,
                              void* d_out, int out_size, void* d_ws, size_t ws_size,
                              hipStream_t stream) {
  (void)in_sizes; (void)n_in; (void)out_size; (void)ws_size;

  const float* x              = (const float*)d_in[0];  // [T, H]
  const float* gate_w         = (const float*)d_in[1];  // [E, H]
  const float* w_gate_up      = (const float*)d_in[2];  // [E, H, 2I]
  const float* w_down         = (const float*)d_in[3];  // [E, I, H]
  const float* shared_gate_up = (const float*)d_in[4];  // [H, 2*SHI]
  const float* shared_down    = (const float*)d_in[5];  // [SHI, H]
  float* out = (float*)d_out;                           // [T, H]

  float* cw     = (float*)d_ws;                  // T*E
  int*   cnt    = (int*)(cw + T_TOK * NEXP);     // E (padded to 16)
  int*   list   = cnt + 16;                      // E*T
  float* wts    = (float*)(list + NEXP * T_TOK); // E*T
  float* act_sh = wts + NEXP * T_TOK;            // T*SHI

  router_kernel<<<T_TOK / 4, 128, 0, stream>>>(x, gate_w, cw);
  compact_kernel<<<NEXP, 32, 0, stream>>>(cw, list, wts, cnt);
  shared_gateup_kernel<<<dim3(T_TOK / BM, SHI / BN), 256, 0, stream>>>(
      x, shared_gate_up, act_sh);
  shared_down_kernel<<<dim3(T_TOK / BM, HID / BN), 256, 0, stream>>>(
      act_sh, shared_down, out);
  routed_fused_kernel<<<dim3(T_TOK / BM, IMOE / ISL, NEXP), 256, 0, stream>>>(
      x, w_gate_up, w_down, list, wts, cnt, out);
}